// Decoder_21930103014212
// MI455X (gfx1250) — compile-verified
//
#include <hip/hip_runtime.h>
#include <hip/hip_bf16.h>
#include <math.h>
#include <stdint.h>

#define BATCH 8
#define NANCH 16384
#define NCLS  81
#define KTOP  100
#define KPAD  112               // 7 tiles of 16 for WMMA
#define NEGK  (-1.0e30f)

typedef __attribute__((ext_vector_type(2))) float v2f;
typedef __attribute__((ext_vector_type(8))) float v8f;

// ---------------------------------------------------------------------------
// Pass 1: per-anchor foreground flag (argmax over C != 0  <=>  any p[c] > p[0])
// ---------------------------------------------------------------------------
__global__ __launch_bounds__(256) void fg_kernel(const float* __restrict__ probs,
                                                 unsigned char* __restrict__ fg) {
  int gid = blockIdx.x * 256 + threadIdx.x;           // over B*N
  if (gid >= BATCH * NANCH) return;
  const float* p = probs + (size_t)gid * NCLS;
  __builtin_prefetch(p + NCLS, 0, 0);                 // global_prefetch_b8
  float p0 = p[0];
  int any = 0;
#pragma unroll 8
  for (int c = 1; c < NCLS; ++c) any |= (p[c] > p0);
  fg[gid] = (unsigned char)any;
}

// ---------------------------------------------------------------------------
// Pass 2: per (b,c) top-100 over N=16384 masked scores + box decode.
// Column staged into LDS with CDNA5 async global->LDS loads.
// ---------------------------------------------------------------------------
__global__ __launch_bounds__(256) void topk_class_kernel(
    const float* __restrict__ probs, const float* __restrict__ rois,
    const float* __restrict__ deltas, const unsigned char* __restrict__ fg,
    float* __restrict__ top_scores, float* __restrict__ top_boxes) {
  extern __shared__ float s[];                        // NANCH floats (64 KB)
  __shared__ float rval[8];
  __shared__ int   ridx[8];
  __shared__ int   selIdx[KTOP];

  const int bc = blockIdx.x;
  const int b  = bc / NCLS;
  const int c  = bc % NCLS;
  const int tid = threadIdx.x;

  // --- async gather of the class-c score column into LDS (L2-resident) ---
  const float* col = probs + ((size_t)b * NANCH) * NCLS + c;
  for (int i = tid; i < NANCH; i += 256) {
    unsigned lds_addr = (unsigned)(uintptr_t)&s[i];
    const float* g = col + (size_t)i * NCLS;
    asm volatile("global_load_async_to_lds_b32 %0, %1, off"
                 :: "v"(lds_addr), "v"(g) : "memory");
  }
  asm volatile("s_wait_asynccnt 0" ::: "memory");
  __syncthreads();

  // --- apply background mask in LDS ---
  const unsigned char* fgb = fg + (size_t)b * NANCH;
  for (int i = tid; i < NANCH; i += 256)
    if (!fgb[i]) s[i] = 0.0f;
  __syncthreads();

  // --- 100 stable max-extractions (value desc, index asc on ties) ---
  for (int k = 0; k < KTOP; ++k) {
    float bv = -1.0f;
    int   bi = NANCH;
    for (int i = tid; i < NANCH; i += 256) {
      float v = s[i];
      if (v > bv) { bv = v; bi = i; }                 // ascending i: ties keep min i
    }
#pragma unroll
    for (int off = 16; off > 0; off >>= 1) {
      float ov = __shfl_down(bv, off, 32);
      int   oi = __shfl_down(bi, off, 32);
      if (ov > bv || (ov == bv && oi < bi)) { bv = ov; bi = oi; }
    }
    if ((tid & 31) == 0) { rval[tid >> 5] = bv; ridx[tid >> 5] = bi; }
    __syncthreads();
    if (tid == 0) {
      float fv = rval[0]; int fi = ridx[0];
#pragma unroll
      for (int w = 1; w < 8; ++w) {
        float ov = rval[w]; int oi = ridx[w];
        if (ov > fv || (ov == fv && oi < fi)) { fv = ov; fi = oi; }
      }
      top_scores[bc * KTOP + k] = fv;
      selIdx[k] = fi;
      s[fi] = -1.0f;                                  // remove (scores are >= 0)
    }
    __syncthreads();
  }

  // --- decode + clip the 100 selected boxes (sparse delta gather) ---
  if (tid < KTOP) {
    int n = selIdx[tid];
    const float* a = rois + ((size_t)b * NANCH + n) * 4;
    float a0 = a[0], a1 = a[1], a2 = a[2], a3 = a[3];
    const float* dl = deltas + (((size_t)b * NANCH + n) * NCLS + c) * 4;
    float d0 = dl[0] * 0.1f, d1 = dl[1] * 0.1f;
    float d2 = dl[2] * 0.2f, d3 = dl[3] * 0.2f;
    float aw = a3 - a1, ah = a2 - a0;
    float acx = a1 + 0.5f * aw, acy = a0 + 0.5f * ah;
    float bw = expf(d3) * aw, bh = expf(d2) * ah;
    float bcx = d1 * aw + acx, bcy = d0 * ah + acy;
    float y1 = bcy - 0.5f * bh, x1 = bcx - 0.5f * bw;
    float y2 = bh + y1, x2 = bw + x1;
    float* ob = top_boxes + (size_t)(bc * KTOP + tid) * 4;
    ob[0] = fminf(fmaxf(y1, 0.0f), 1.0f);
    ob[1] = fminf(fmaxf(x1, 0.0f), 1.0f);
    ob[2] = fminf(fmaxf(y2, 0.0f), 1.0f);
    ob[3] = fminf(fmaxf(x2, 0.0f), 1.0f);
  }
}

// ---------------------------------------------------------------------------
// Pass 3: per (b,c) greedy NMS over 100 boxes.
// WMMA f32 16x16x4 builds the (area_i + area_j) outer-sum tiles; VALU min/max
// computes the intersections (co-executes with the XDL pipe); suppression
// bitmatrix in LDS; exact sequential greedy on one lane; emits keyed scores.
// ---------------------------------------------------------------------------
__global__ __launch_bounds__(128) void nms_kernel(const float* __restrict__ top_boxes,
                                                  const float* __restrict__ top_scores,
                                                  float* __restrict__ keyed) {
  __shared__ float by1[KPAD], bx1[KPAD], by2[KPAD], bx2[KPAD], area[KPAD];
  __shared__ unsigned supp[KPAD][4];
  __shared__ unsigned keepw[4];
  const int bc = blockIdx.x;
  const int tid = threadIdx.x;

  for (int k = tid; k < KPAD; k += 128) {
    float y1 = 0.f, x1 = 0.f, y2 = 0.f, x2 = 0.f;
    if (k < KTOP) {
      const float* p = top_boxes + (size_t)(bc * KTOP + k) * 4;
      y1 = p[0]; x1 = p[1]; y2 = p[2]; x2 = p[3];
    }
    by1[k] = y1; bx1[k] = x1; by2[k] = y2; bx2[k] = x2;
    area[k] = (y2 - y1) * (x2 - x1);
    supp[k][0] = 0u; supp[k][1] = 0u; supp[k][2] = 0u; supp[k][3] = 0u;
  }
  __syncthreads();

  const int wave = tid >> 5;
  const int lane = tid & 31;
  const bool lo  = lane < 16;
  const int  l   = lane & 15;
  for (int t = wave; t < 49; t += 4) {                // uniform per wave: EXEC all ones
    int ti = t / 7, tj = t % 7;
    // A (16x4): col0 = area_i, col1 = 1 ;  B (4x16): row0 = 1, row1 = area_j
    v2f A, Bm;
    A.x  = lo ? area[ti * 16 + l] : 0.0f;             // K=0 (lo) / K=2 (hi)
    A.y  = lo ? 1.0f              : 0.0f;             // K=1 (lo) / K=3 (hi)
    Bm.x = lo ? 1.0f              : 0.0f;
    Bm.y = lo ? area[tj * 16 + l] : 0.0f;
    v8f Cz = {};
    v8f D = __builtin_amdgcn_wmma_f32_16x16x4_f32(
        false, A, false, Bm, (short)0, Cz, false, false);
#pragma unroll
    for (int v = 0; v < 8; ++v) {
      int i = ti * 16 + v + (lo ? 0 : 8);
      int j = tj * 16 + l;
      if (i < KTOP && j < KTOP && j > i) {
        float iy1 = fmaxf(by1[i], by1[j]);
        float ix1 = fmaxf(bx1[i], bx1[j]);
        float iy2 = fminf(by2[i], by2[j]);
        float ix2 = fminf(bx2[i], bx2[j]);
        float inter = fmaxf(iy2 - iy1, 0.0f) * fmaxf(ix2 - ix1, 0.0f);
        float uni = D[v] - inter;                     // area_i + area_j - inter
        float iou = inter / fmaxf(uni, 1e-12f);
        if (iou > 0.5f) atomicOr(&supp[i][j >> 5], 1u << (j & 31));
      }
    }
  }
  __syncthreads();

  if (tid == 0) {
    unsigned k0 = ~0u, k1 = ~0u, k2 = ~0u, k3 = ~0u;
    for (int i = 0; i < KTOP; ++i) {
      unsigned w = (i < 32) ? k0 : (i < 64) ? k1 : (i < 96) ? k2 : k3;
      if ((w >> (i & 31)) & 1u) {                     // box i still kept
        k0 &= ~supp[i][0]; k1 &= ~supp[i][1];
        k2 &= ~supp[i][2]; k3 &= ~supp[i][3];
      }
    }
    keepw[0] = k0; keepw[1] = k1; keepw[2] = k2; keepw[3] = k3;
  }
  __syncthreads();

  if (tid < KTOP) {
    bool kp = (keepw[tid >> 5] >> (tid & 31)) & 1u;
    float sc = top_scores[bc * KTOP + tid];
    keyed[bc * KTOP + tid] = kp ? sc : NEGK;
  }
}

// ---------------------------------------------------------------------------
// Pass 4: per batch, top-100 over C*K = 8100 keyed entries; write outputs.
// ---------------------------------------------------------------------------
__global__ __launch_bounds__(256) void final_topk_kernel(
    const float* __restrict__ keyed, const float* __restrict__ top_scores,
    const float* __restrict__ top_boxes, float* __restrict__ out) {
  __shared__ float s[NCLS * KTOP];                    // 8100 floats
  __shared__ float rval[8];
  __shared__ int   ridx[8];
  __shared__ int   selIdx[KTOP];
  __shared__ float selKey[KTOP];
  const int b = blockIdx.x;
  const int tid = threadIdx.x;
  const int M = NCLS * KTOP;

  const float* kin = keyed + (size_t)b * M;
  for (int i = tid; i < M; i += 256) s[i] = kin[i];
  __syncthreads();

  for (int k = 0; k < KTOP; ++k) {
    float bv = -INFINITY;
    int   bi = M;
    for (int i = tid; i < M; i += 256) {
      float v = s[i];
      if (v > bv) { bv = v; bi = i; }
    }
#pragma unroll
    for (int off = 16; off > 0; off >>= 1) {
      float ov = __shfl_down(bv, off, 32);
      int   oi = __shfl_down(bi, off, 32);
      if (ov > bv || (ov == bv && oi < bi)) { bv = ov; bi = oi; }
    }
    if ((tid & 31) == 0) { rval[tid >> 5] = bv; ridx[tid >> 5] = bi; }
    __syncthreads();
    if (tid == 0) {
      float fv = rval[0]; int fi = ridx[0];
#pragma unroll
      for (int w = 1; w < 8; ++w) {
        float ov = rval[w]; int oi = ridx[w];
        if (ov > fv || (ov == fv && oi < fi)) { fv = ov; fi = oi; }
      }
      selKey[k] = fv; selIdx[k] = fi;
      s[fi] = -1.0e31f;                               // below NEG: extracted
    }
    __syncthreads();
  }

  if (tid < KTOP) {
    int f = selIdx[tid];
    bool valid = selKey[tid] > (NEGK * 0.5f);
    int c = f / KTOP;
    const float* pb = top_boxes + (size_t)(b * M + f) * 4;
    float b0 = pb[0], b1 = pb[1], b2 = pb[2], b3 = pb[3];
    float sc = top_scores[(size_t)b * M + f];
    float cls = (float)c;
    if (!valid) { b0 = b1 = b2 = b3 = 0.0f; cls = 0.0f; sc = 0.0f; }
    float* ob = out + (size_t)(b * KTOP + tid) * 4;   // boxes region [0, 3200)
    ob[0] = b0; ob[1] = b1; ob[2] = b2; ob[3] = b3;
    out[BATCH * KTOP * 4 + b * KTOP + tid] = cls;     // classes at 3200
    out[BATCH * KTOP * 5 + b * KTOP + tid] = sc;      // scores at 4000
  }
}

// ---------------------------------------------------------------------------
extern "C" void kernel_launch(void* const* d_in, const int* in_sizes, int n_in,
                              void* d_out, int out_size, void* d_ws, size_t ws_size,
                              hipStream_t stream) {
  const float* roi    = (const float*)d_in[0];   // (8,16384,4)
  const float* deltas = (const float*)d_in[1];   // (8,16384,324)
  const float* probs  = (const float*)d_in[2];   // (8,16384,81)
  float* out = (float*)d_out;                    // 4800 floats

  char* ws = (char*)d_ws;
  unsigned char* fg  = (unsigned char*)ws;                       // 131072 B
  float* top_scores  = (float*)(ws + 131072);                    // 259200 B
  float* top_boxes   = (float*)(ws + 131072 + 259200);           // 1036800 B
  float* keyed       = (float*)(ws + 131072 + 259200 + 1036800); // 259200 B

  fg_kernel<<<(BATCH * NANCH + 255) / 256, 256, 0, stream>>>(probs, fg);
  topk_class_kernel<<<BATCH * NCLS, 256, NANCH * sizeof(float), stream>>>(
      probs, roi, deltas, fg, top_scores, top_boxes);
  nms_kernel<<<BATCH * NCLS, 128, 0, stream>>>(top_boxes, top_scores, keyed);
  final_topk_kernel<<<BATCH, 256, 0, stream>>>(keyed, top_scores, top_boxes, out);
}